// Crf_layer_16999480558330
// MI455X (gfx1250) — compile-verified
//
#include <hip/hip_runtime.h>
#include <hip/hip_bf16.h>

#define Bn 256
#define Tn 512
#define Dn 256
#define Ln 32

typedef __attribute__((ext_vector_type(2))) float v2f;
typedef __attribute__((ext_vector_type(8))) float v8f;

// ---------------------------------------------------------------------------
// Kernel 1: projection GEMM  out[b,t,l] = sum_d inputs[b,t,d] * W[d,l]
// M = B*T = 131072, K = 256, N = 32.  f32 WMMA 16x16x4.
// Block = 256 threads (8 waves). Each wave: 16 rows x 32 cols.
// W is staged into LDS pre-swizzled into WMMA B-fragment order so each
// lane's operand pair is a single ds_load_b64 (no repacking movs).
// ---------------------------------------------------------------------------
__global__ __launch_bounds__(256) void crf_gemm_kernel(
    const float* __restrict__ inputs,   // (B*T, D)
    const float* __restrict__ W,        // (D, L)
    float* __restrict__ outM)           // (B*T, L)
{
    // Fragment-order B operands: index = kkq*32 + lane, kkq = k-step (K/4).
    __shared__ v2f sB0[(Dn / 4) * 32];   // cols 0..15,  16 KB
    __shared__ v2f sB1[(Dn / 4) * 32];   // cols 16..31, 16 KB

    const int tid = threadIdx.x;

    // Pre-swizzle W into fragment order.
    // Fragment layout (16x16x4 f32 B): lane l, comp j: K = 4*kkq + 2*(l/16)+j,
    // N = l%16 (+16 for the second tile).
    for (int idx = tid; idx < (Dn / 4) * 32; idx += 256) {
        const int kkq  = idx >> 5;
        const int l    = idx & 31;
        const int half = l >> 4;
        const int n    = l & 15;
        const int k0   = kkq * 4 + 2 * half;
        v2f b0, b1;
        b0.x = W[k0 * Ln + n];
        b0.y = W[(k0 + 1) * Ln + n];
        b1.x = W[k0 * Ln + n + 16];
        b1.y = W[(k0 + 1) * Ln + n + 16];
        sB0[idx] = b0;
        sB1[idx] = b1;
    }
    __syncthreads();

    const int wave = tid >> 5;
    const int lane = tid & 31;
    const int half = lane >> 4;     // 0 or 1
    const int l16  = lane & 15;

    const int row_base = (blockIdx.x * 8 + wave) * 16;   // 16 rows per wave

    const float* arow = inputs + (size_t)(row_base + l16) * Dn + 2 * half;
    const v2f* pB0 = sB0 + lane;
    const v2f* pB1 = sB1 + lane;

    v8f c0 = {};   // cols 0..15
    v8f c1 = {};   // cols 16..31

    #pragma unroll 4
    for (int kkq = 0; kkq < Dn / 4; ++kkq) {
        // A fragment: lane l: M = l%16, K = 4*kkq + 2*(l/16) + j
        v2f a  = *(const v2f*)(arow + kkq * 4);
        v2f b0 = pB0[kkq * 32];
        v2f b1 = pB1[kkq * 32];

        c0 = __builtin_amdgcn_wmma_f32_16x16x4_f32(false, a, false, b0,
                                                   (short)0, c0, false, false);
        c1 = __builtin_amdgcn_wmma_f32_16x16x4_f32(false, a, false, b1,
                                                   (short)0, c1, false, false);
    }

    // D layout: VGPR v, lanes 0-15 -> M=v, lanes 16-31 -> M=v+8, N=lane%16
    float* orow = outM + (size_t)row_base * Ln;
    #pragma unroll
    for (int v = 0; v < 8; ++v) {
        orow[(v + 8 * half) * Ln + l16]      = c0[v];
        orow[(v + 8 * half) * Ln + l16 + 16] = c1[v];
    }
}

// ---------------------------------------------------------------------------
// Kernel 2: per-batch CRF forward (logsumexp) + Viterbi + backtrace.
// One wave (32 lanes) per batch element: lane j <-> state j.
// ---------------------------------------------------------------------------
__global__ __launch_bounds__(32) void crf_seq_kernel(
    const float* __restrict__ outM,     // (B*T, L)
    const float* __restrict__ trans,    // (L, L)
    const int*   __restrict__ targets,  // (B, T)
    const int*   __restrict__ lens,     // (B,)
    float* __restrict__ ll,             // (B,)  log-likelihood
    float* __restrict__ preds)          // (B*T) predictions (stored as float)
{
    __shared__ unsigned char bp[(Tn - 1) * Ln];   // backpointers, 16 KB

    const int b    = blockIdx.x;
    const int lane = threadIdx.x;      // 0..31
    const int len  = lens[b];

    // transition column for this lane: col[i] = trans[i][lane]
    float col[Ln];
    #pragma unroll
    for (int i = 0; i < Ln; ++i)
        col[i] = trans[i * Ln + lane];

    const float* xrow = outM + (size_t)b * Tn * Ln;

    float alpha  = xrow[lane];     // logsumexp alpha
    float valpha = alpha;          // viterbi alpha

    for (int t = 1; t < len; ++t) {
        const float x = xrow[t * Ln + lane];

        float s[Ln];
        float m  = -__builtin_inff();
        float vm = -__builtin_inff();
        int   arg = 0;
        #pragma unroll
        for (int i = 0; i < Ln; ++i) {
            const float ai  = __shfl(alpha,  i, 32);
            const float vai = __shfl(valpha, i, 32);
            s[i] = ai + col[i];
            m = fmaxf(m, s[i]);
            const float vs = vai + col[i];
            if (vs > vm) { vm = vs; arg = i; }   // keep first max (argmax)
        }
        float sum = 0.0f;
        #pragma unroll
        for (int i = 0; i < Ln; ++i)
            sum += __expf(s[i] - m);

        alpha  = m + __logf(sum) + x;
        valpha = vm + x;
        bp[(t - 1) * Ln + lane] = (unsigned char)arg;
    }
    __syncthreads();

    // log_norm = logsumexp over lanes of alpha
    float am = alpha;
    #pragma unroll
    for (int off = 16; off > 0; off >>= 1)
        am = fmaxf(am, __shfl_xor(am, off, 32));
    float ae = __expf(alpha - am);
    #pragma unroll
    for (int off = 16; off > 0; off >>= 1)
        ae += __shfl_xor(ae, off, 32);
    const float log_norm = am + __logf(ae);

    // last_tag = argmax over lanes of valpha (lowest index on ties)
    float vv = valpha;
    int   vi = lane;
    #pragma unroll
    for (int off = 16; off > 0; off >>= 1) {
        const float ov = __shfl_xor(vv, off, 32);
        const int   oi = __shfl_xor(vi, off, 32);
        if (ov > vv || (ov == vv && oi < vi)) { vv = ov; vi = oi; }
    }
    const int last_tag = vi;

    // unary / binary scores (lane-parallel over t, then reduce)
    const int* tgt = targets + (size_t)b * Tn;
    float us = 0.0f, bs = 0.0f;
    for (int t = lane; t < Tn; t += 32) {
        if (t < len) {
            us += xrow[t * Ln + tgt[t]];
            if (t >= 1) bs += trans[tgt[t - 1] * Ln + tgt[t]];
        }
    }
    #pragma unroll
    for (int off = 16; off > 0; off >>= 1) {
        us += __shfl_xor(us, off, 32);
        bs += __shfl_xor(bs, off, 32);
    }

    if (lane == 0)
        ll[b] = us + bs - log_norm;

    // predictions: zero-fill masked tail, then backtrace valid prefix
    float* pr = preds + (size_t)b * Tn;
    for (int t = lane; t < Tn; t += 32)
        if (t >= len) pr[t] = 0.0f;

    if (lane == 0) {
        int tag = last_tag;
        pr[len - 1] = (float)tag;
        for (int k = len - 2; k >= 0; --k) {
            tag = bp[k * Ln + tag];
            pr[k] = (float)tag;
        }
    }
}

// ---------------------------------------------------------------------------
// Kernel 3: loss = mean(-ll); copy trans to output tail.
// ---------------------------------------------------------------------------
__global__ __launch_bounds__(256) void crf_finalize_kernel(
    const float* __restrict__ ll,
    const float* __restrict__ trans,
    float* __restrict__ out)
{
    __shared__ float red[256];
    const int tid = threadIdx.x;
    red[tid] = ll[tid];
    __syncthreads();
    #pragma unroll
    for (int s = 128; s > 0; s >>= 1) {
        if (tid < s) red[tid] += red[tid + s];
        __syncthreads();
    }
    if (tid == 0)
        out[0] = -red[0] / (float)Bn;

    #pragma unroll
    for (int i = 0; i < Ln * Ln / 256; ++i)
        out[1 + Bn * Tn + tid + i * 256] = trans[tid + i * 256];
}

// ---------------------------------------------------------------------------
extern "C" void kernel_launch(void* const* d_in, const int* in_sizes, int n_in,
                              void* d_out, int out_size, void* d_ws, size_t ws_size,
                              hipStream_t stream) {
    const float* inputs  = (const float*)d_in[0];   // (B,T,D)
    const float* W       = (const float*)d_in[1];   // (D,L)
    const float* trans   = (const float*)d_in[2];   // (L,L)
    const int*   targets = (const int*)d_in[3];     // (B,T)
    const int*   lens    = (const int*)d_in[4];     // (B,)

    float* outM = (float*)d_ws;                                   // B*T*L f32
    float* ll   = (float*)d_ws + (size_t)Bn * Tn * Ln;            // B f32

    float* out   = (float*)d_out;
    float* preds = out + 1;                                       // B*T

    // GEMM: 131072 rows / (8 waves * 16 rows) = 1024 blocks
    crf_gemm_kernel<<<dim3((Bn * Tn) / 128), dim3(256), 0, stream>>>(inputs, W, outM);

    // CRF recurrence: one wave per batch element
    crf_seq_kernel<<<dim3(Bn), dim3(32), 0, stream>>>(outM, trans, targets, lens, ll, preds);

    // Loss + trans passthrough
    crf_finalize_kernel<<<dim3(1), dim3(256), 0, stream>>>(ll, trans, out);
}